// GenerativeModel_59502476918796
// MI455X (gfx1250) — compile-verified
//
#include <hip/hip_runtime.h>
#include <math.h>

typedef __attribute__((ext_vector_type(2))) float        v2f;
typedef __attribute__((ext_vector_type(8))) float        v8f;
typedef __attribute__((ext_vector_type(4))) unsigned int u32x4;
typedef __attribute__((ext_vector_type(8))) unsigned int u32x8;

#define EPS_PRIOR 0.001f
#define CLAMP_C   0.0001f
#define DECAY     0.1f
#define LOG2PI    1.8378770664093453f
#define CHUNK     32          // rows of d/x staged per TDM tile

// ---------------- TDM: 2D f32 tile  global -> LDS  (ISA 08, D# groups) ------
// group0: [1:0]=count=1, [63:32]=lds_addr(bytes), [120:64]=global_addr(bytes),
//         [127:126]=type=2
// group1: [17:16]=data_size=2 (4B), [79:48]=tensor_dim0, [111:80]=tensor_dim1,
//         [127:112]=tile_dim0, [143:128]=tile_dim1, [207:160]=tensor_dim0_stride
__device__ inline void tdm_load_2d_f32(unsigned lds_byte_off,
                                       const float* gptr,
                                       unsigned rows, unsigned cols,
                                       unsigned stride_elems) {
    unsigned long long ga = (unsigned long long)(uintptr_t)gptr;
    u32x4 g0;
    g0.x = 1u;                                            // count = 1
    g0.y = lds_byte_off;                                  // lds_addr
    g0.z = (unsigned)(ga & 0xffffffffull);                // global_addr lo
    g0.w = (unsigned)((ga >> 32) & 0x1ffffffull) | (2u << 30); // ga hi | type=2
    u32x8 g1;
    g1[0] = (2u << 16);                                   // data_size = 4 bytes
    g1[1] = (cols & 0xffffu) << 16;                       // tensor_dim0 lo16
    g1[2] = ((cols >> 16) & 0xffffu) | ((rows & 0xffffu) << 16); // td0 hi | td1 lo
    g1[3] = ((rows >> 16) & 0xffffu) | ((cols & 0xffffu) << 16); // td1 hi | tile_dim0
    g1[4] = (rows & 0xffffu);                             // tile_dim1 (tile_dim2 = 0)
    g1[5] = stride_elems;                                 // tensor_dim0_stride lo32
    g1[6] = 0u;
    g1[7] = 0u;
    u32x4 gz = {0u, 0u, 0u, 0u};                          // 2D: groups 2/3 zero
    asm volatile("tensor_load_to_lds %0, %1, %2, %3"
                 :: "s"(g0), "s"(g1), "s"(gz), "s"(gz)
                 : "memory");
}

// ---------------- kernel 1: sequential decay scan  d[i][k] -------------------
__global__ void prep_d_kernel(const float* __restrict__ x, float* __restrict__ d,
                              int n, int Dx) {
    int i = blockIdx.x * blockDim.x + threadIdx.x;
    if (i >= n) return;
    const float* xr = x + (size_t)i * Dx;
    float*       dr = d + (size_t)i * Dx;
    float acc = 0.f;
    for (int k = 0; k < Dx; ++k) {
        acc = (acc + xr[k]) * DECAY;
        dr[k] = acc;
    }
}

// ---------------- kernel 2: S0 = sum(x[:,0]) --------------------------------
__global__ void s0_kernel(const float* __restrict__ x, float* __restrict__ s0,
                          int n, int Dx) {
    __shared__ float red[256];
    float acc = 0.f;
    for (int i = threadIdx.x; i < n; i += 256) acc += x[(size_t)i * Dx];
    red[threadIdx.x] = acc;
    __syncthreads();
    for (int off = 128; off > 0; off >>= 1) {
        if (threadIdx.x < (unsigned)off) red[threadIdx.x] += red[threadIdx.x + off];
        __syncthreads();
    }
    if (threadIdx.x == 0) s0[0] = red[0];
}

// ---------------- kernel 3: up = z @ theta via V_WMMA_F32_16X16X4_F32 -------
__global__ void up_wmma_kernel(const float* __restrict__ z,
                               const float* __restrict__ th,
                               float* __restrict__ up,
                               int M, int K, int N) {
    int lane   = threadIdx.x & 31;
    int wave   = threadIdx.x >> 5;
    int tilesN = (N + 15) >> 4;
    int tilesM = (M + 15) >> 4;
    int t = blockIdx.x * (blockDim.x >> 5) + wave;   // wave-uniform
    if (t >= tilesM * tilesN) return;                // uniform branch, EXEC full
    int tm = t / tilesN;
    int tn = t - tm * tilesN;

    int mrow = tm * 16 + (lane & 15);
    int col  = tn * 16 + (lane & 15);
    int k0   = (lane < 16) ? 0 : 2;
    int k1   = k0 + 1;

    v2f a, b;
    a.x = (mrow < M && k0 < K) ? z[(size_t)mrow * K + k0] : 0.f;
    a.y = (mrow < M && k1 < K) ? z[(size_t)mrow * K + k1] : 0.f;
    b.x = (col  < N && k0 < K) ? th[(size_t)k0 * N + col] : 0.f;
    b.y = (col  < N && k1 < K) ? th[(size_t)k1 * N + col] : 0.f;

    v8f c = {};
    c = __builtin_amdgcn_wmma_f32_16x16x4_f32(false, a, false, b,
                                              (short)0, c, false, false);

    if (col < N) {
        int rbase = tm * 16 + ((lane & 16) ? 8 : 0);
        size_t base = (size_t)rbase * N + col;
#pragma unroll
        for (int r = 0; r < 8; ++r) {
            if (rbase + r < M) up[base] = c[r];
            base += N;
        }
    }
}

// ---------------- kernel 4: O(n^2 * Dx) log-likelihood sum ------------------
// block = (j, dz); thread tid holds k = tid; i-loop over TDM-staged LDS chunks.
// LDS layout (dynamic): buffer b in {0,1}: d rows at b*2*bufElems,
//                                          x rows at b*2*bufElems + bufElems.
__global__ void main_kernel(const float* __restrict__ z,
                            const float* __restrict__ x,
                            const float* __restrict__ d,
                            const float* __restrict__ up,
                            const float* __restrict__ s0p,
                            float* __restrict__ out,
                            int n, int P, int Dz, int Dx) {
    extern __shared__ float smem[];
    __shared__ float red[128];
    const int j   = blockIdx.x;
    const int dz  = blockIdx.y;
    const int tid = threadIdx.x;
    const bool wave0  = (tid >> 5) == 0;             // wave-uniform
    const bool active = tid < Dx - 1;

    const float* urow    = up + ((size_t)j * P + dz) * Dx;
    const size_t bufElems = (size_t)CHUNK * Dx;
    const unsigned bufBytes = (unsigned)(bufElems * sizeof(float));
    const unsigned lds0 = (unsigned)(uintptr_t)smem; // LDS byte offset
    const int nchunks = (n + CHUNK - 1) / CHUNK;

    // prologue: stage chunk 0 into buffer 0
    if (wave0) {
        unsigned rows0 = (unsigned)min(CHUNK, n);
        tdm_load_2d_f32(lds0,            d, rows0, (unsigned)Dx, (unsigned)Dx);
        tdm_load_2d_f32(lds0 + bufBytes, x, rows0, (unsigned)Dx, (unsigned)Dx);
        __builtin_amdgcn_s_wait_tensorcnt(0);
    }
    __syncthreads();

    float u = 0.f;
    if (active) u = urow[tid + 1];
    float acc = 0.f;

    for (int c = 0; c < nchunks; ++c) {
        // issue chunk c+1 into the other buffer while computing chunk c
        if (wave0 && (c + 1) < nchunks) {
            int i0 = (c + 1) * CHUNK;
            unsigned rows = (unsigned)min(CHUNK, n - i0);
            unsigned dst  = lds0 + (unsigned)((c + 1) & 1) * 2u * bufBytes;
            tdm_load_2d_f32(dst,            d + (size_t)i0 * Dx, rows, Dx, Dx);
            tdm_load_2d_f32(dst + bufBytes, x + (size_t)i0 * Dx, rows, Dx, Dx);
        }

        const float* bufD = smem + (size_t)(c & 1) * 2 * bufElems;
        const float* bufX = bufD + bufElems;
        const int rows = min(CHUNK, n - c * CHUNK);
        if (active) {
            for (int ii = 0; ii < rows; ++ii) {
                float dv = bufD[(size_t)ii * Dx + tid];      // d_prev[i][k]
                float xv = bufX[(size_t)ii * Dx + tid + 1];  // xr[i][k]
                float s  = dv + u;
                float e  = __expf(-s);
                float num = (xv != 0.f) ? (1.f - CLAMP_C - CLAMP_C * e)
                                        : (CLAMP_C + (1.f + CLAMP_C) * e);
                acc += __logf(num) - __logf(1.f + e);
            }
        }
        __syncthreads();                                // reads of buffer c&1 done
        if (wave0 && (c + 1) < nchunks) {
            __builtin_amdgcn_s_wait_tensorcnt(0);       // chunk c+1 landed
        }
        __syncthreads();
    }

    red[tid] = acc;
    __syncthreads();
    for (int off = 64; off > 0; off >>= 1) {
        if (tid < off) red[tid] += red[tid + off];
        __syncthreads();
    }

    if (tid == 0) {
        float S0 = s0p[0];
        float u0 = urow[0];
        float p0 = 1.f / (1.f + __expf(-u0));
        float lp0 = __logf(p0 - CLAMP_C) * S0
                  + __logf(1.f - p0 + CLAMP_C) * ((float)n - S0);

        float z0 = z[(((size_t)j * P) + dz) * Dz + 0];
        float z1 = z[(((size_t)j * P) + dz) * Dz + 1];
        float t0 = z0 / EPS_PRIOR;
        float t1 = z1 * __expf(-z0 * 0.25f);
        float lp = -0.5f * t0 * t0 - __logf(EPS_PRIOR) - 0.5f * LOG2PI
                   -0.5f * t1 * t1 - (z0 * 0.25f)      - 0.5f * LOG2PI;

        out[(size_t)j * P + dz] = lp + lp0 + red[0];
    }
}

// ---------------------------------------------------------------------------
extern "C" void kernel_launch(void* const* d_in, const int* in_sizes, int n_in,
                              void* d_out, int out_size, void* d_ws, size_t ws_size,
                              hipStream_t stream) {
    const float* z  = (const float*)d_in[0];   // (n, P, Dz)
    const float* x  = (const float*)d_in[1];   // (n, Dx)
    const float* th = (const float*)d_in[2];   // (Dz, Dx)

    const int P  = 2;
    const int Dz = 2;
    int n  = in_sizes[0] / (P * Dz);           // 512
    int Dx = in_sizes[1] / n;                  // 128

    float* ws   = (float*)d_ws;
    float* s0   = ws;                          // 1 float (64B-padded slot)
    float* dmat = ws + 16;                     // n*Dx floats
    float* up   = dmat + (size_t)n * Dx;       // (n*P)*Dx floats
    float* out  = (float*)d_out;

    prep_d_kernel<<<(n + 255) / 256, 256, 0, stream>>>(x, dmat, n, Dx);
    s0_kernel<<<1, 256, 0, stream>>>(x, s0, n, Dx);

    int M = n * P;
    int tiles = ((M + 15) / 16) * ((Dx + 15) / 16);
    up_wmma_kernel<<<(tiles + 7) / 8, 256, 0, stream>>>(z, th, up, M, Dz, Dx);

    size_t smemBytes = (size_t)2 * 2 * CHUNK * Dx * sizeof(float); // 64 KB @ Dx=128
    dim3 grid(n, P);
    main_kernel<<<grid, 128, smemBytes, stream>>>(z, x, dmat, up, s0, out,
                                                  n, P, Dz, Dx);
}